// WPATask_24704651886645
// MI455X (gfx1250) — compile-verified
//
#include <hip/hip_runtime.h>
#include <hip/hip_bf16.h>
#include <math.h>

#define EMBED  768
#define W1COLS 1536
#define TB     8          // token rows per score block

typedef __attribute__((ext_vector_type(2))) float v2f;
typedef __attribute__((ext_vector_type(8))) float v8f;

static __device__ __forceinline__ v8f wmma_f32(v2f a, v2f b, v8f c) {
    return __builtin_amdgcn_wmma_f32_16x16x4_f32(
        /*neg_a=*/false, a, /*neg_b=*/false, b,
        /*c_mod=*/(short)0, c, /*reuse_a=*/false, /*reuse_b=*/false);
}

// ---------------------------------------------------------------------------
// Projection GEMM:  Out[m,h] = sum_e A[m,e] * W1[h, wcol_off + e]  (+ b1[h])
// One wave computes a 32x32 output block (2x2 tiles of 16x16) with
// V_WMMA_F32_16X16X4_F32: per K-step, 2 A-frag + 2 B-frag loads feed 4 wmma.
//
// f32 WMMA fragment layouts (CDNA5 ISA 7.12.2):
//   A (16x4):  lane&15 = M ; VGPR0 = K=2*(lane>>4), VGPR1 = K=2*(lane>>4)+1
//   B (4x16):  lane&15 = N ; same K striping
//   C/D:       lane&15 = N ; VGPR r = row r + 8*(lane>>4)
// ---------------------------------------------------------------------------
__global__ __launch_bounds__(32)
void WPATask_proj_gemm_wmma(const float* __restrict__ A,
                            const float* __restrict__ W1,
                            const float* __restrict__ b1,
                            float* __restrict__ Out,
                            int M, int wcol_off, int add_bias)
{
    const int lane = threadIdx.x & 31;
    const int half = lane >> 4;          // 0: lanes 0-15, 1: lanes 16-31
    const int l16  = lane & 15;

    const int n0 = blockIdx.x * 32;      // h block  (0..736, step 32)
    const int m0 = blockIdx.y * 32;      // m block

    int ma0 = m0 + l16;      if (ma0 >= M) ma0 = M - 1;   // clamp loads;
    int ma1 = m0 + 16 + l16; if (ma1 >= M) ma1 = M - 1;   // bad rows dropped at store

    const float* __restrict__ A0 = A  + (size_t)ma0 * EMBED;
    const float* __restrict__ A1 = A  + (size_t)ma1 * EMBED;
    const float* __restrict__ W0 = W1 + (size_t)(n0 + l16)      * W1COLS + wcol_off;
    const float* __restrict__ W1r= W1 + (size_t)(n0 + 16 + l16) * W1COLS + wcol_off;

    v8f c00 = {}, c01 = {}, c10 = {}, c11 = {};
#pragma unroll 4
    for (int k = 0; k < EMBED; k += 4) {
        const int kk = k + 2 * half;
        v2f a0 = *(const v2f*)(A0  + kk);
        v2f a1 = *(const v2f*)(A1  + kk);
        v2f b0 = *(const v2f*)(W0  + kk);
        v2f b1v= *(const v2f*)(W1r + kk);
        c00 = wmma_f32(a0, b0,  c00);
        c01 = wmma_f32(a0, b1v, c01);
        c10 = wmma_f32(a1, b0,  c10);
        c11 = wmma_f32(a1, b1v, c11);
    }

    const float bias0 = add_bias ? b1[n0 + l16]      : 0.0f;
    const float bias1 = add_bias ? b1[n0 + 16 + l16] : 0.0f;
    const int nA = n0 + l16;
    const int nB = n0 + 16 + l16;

    if (m0 + 32 <= M) {                  // full block: straight-line stores
#pragma unroll
        for (int r = 0; r < 8; ++r) {
            const int m = m0 + r + 8 * half;
            Out[(size_t)m        * EMBED + nA] = c00[r] + bias0;
            Out[(size_t)m        * EMBED + nB] = c01[r] + bias1;
            Out[(size_t)(m + 16) * EMBED + nA] = c10[r] + bias0;
            Out[(size_t)(m + 16) * EMBED + nB] = c11[r] + bias1;
        }
    } else {                             // ragged tail (patch GEMM last block)
#pragma unroll
        for (int r = 0; r < 8; ++r) {
            const int m = m0 + r + 8 * half;
            if (m < M) {
                Out[(size_t)m * EMBED + nA] = c00[r] + bias0;
                Out[(size_t)m * EMBED + nB] = c01[r] + bias1;
            }
            if (m + 16 < M) {
                Out[(size_t)(m + 16) * EMBED + nA] = c10[r] + bias0;
                Out[(size_t)(m + 16) * EMBED + nB] = c11[r] + bias1;
            }
        }
    }
}

// ---------------------------------------------------------------------------
// Score kernel: one block per (b, group of 8 t's).
//   scores[b,t,p] = sigmoid( sum_h relu(tokp[bt,h] + patp[bp,h]) * w2[h] + b2 )
// 8 token rows (+ w2) staged in LDS (27 KB of the 320 KB WGP pool); each
// patch row is then read once per block instead of once per t (8x less L2
// traffic: ~616 MB -> ~77 MB, keeping the stage VALU-bound). float4 h-loop,
// global_prefetch on the next patch row.
// ---------------------------------------------------------------------------
__global__ __launch_bounds__(256)
void WPATask_score_kernel(const float* __restrict__ tokp,  // 1024 x 768 (b1 folded)
                          const float* __restrict__ patp,  //  392 x 768
                          const float* __restrict__ W2,    //  768
                          const float* __restrict__ b2,    //  1
                          float* __restrict__ out)         // (2,512,196)
{
    __shared__ float s_tok[TB * EMBED];
    __shared__ float s_w2[EMBED];

    const int blk = blockIdx.x;            // 0..127
    const int b   = blk >> 6;              // 64 groups per batch
    const int t0  = (blk & 63) * TB;
    const int bt0 = b * 512 + t0;
    const int tid = threadIdx.x;

    for (int i = tid; i < TB * EMBED; i += 256)
        s_tok[i] = tokp[(size_t)bt0 * EMBED + i];   // 8 consecutive rows
    for (int i = tid; i < EMBED; i += 256)
        s_w2[i] = W2[i];
    __syncthreads();

    const float bb = b2[0];
    for (int pair = tid; pair < TB * 196; pair += 256) {
        const int p  = pair >> 3;          // patch index
        const int tl = pair & (TB - 1);    // token row within group

        const int np = pair + 256;         // prefetch next patch row
        if (np < TB * 196)
            __builtin_prefetch(patp + (size_t)(b * 196 + (np >> 3)) * EMBED, 0, 0);

        const float* __restrict__ prow = patp + (size_t)(b * 196 + p) * EMBED;
        const float* __restrict__ trow = s_tok + tl * EMBED;

        float acc = 0.0f;
#pragma unroll 4
        for (int h = 0; h < EMBED; h += 4) {
            const float4 pv = *(const float4*)(prow + h);
            const float4 tv = *(const float4*)(trow + h);
            const float4 wv = *(const float4*)(s_w2 + h);
            acc = fmaf(fmaxf(tv.x + pv.x, 0.0f), wv.x, acc);
            acc = fmaf(fmaxf(tv.y + pv.y, 0.0f), wv.y, acc);
            acc = fmaf(fmaxf(tv.z + pv.z, 0.0f), wv.z, acc);
            acc = fmaf(fmaxf(tv.w + pv.w, 0.0f), wv.w, acc);
        }
        const float s = acc + bb;
        out[(size_t)(bt0 + tl) * 196 + p] = 1.0f / (1.0f + expf(-s));
    }
}

// ---------------------------------------------------------------------------
// Launcher
//   d_in: 0=token_embeds (2,512,768) 1=patch_embeds (2,196,768)
//         2=W1 (768,1536) 3=b1 (768) 4=W2 (1,768) 5=b2 (1)
//   d_out: (2,512,196) f32
//   d_ws:  tok_proj (1024*768 f32) | pat_proj (392*768 f32)  ~= 4.35 MB
// ---------------------------------------------------------------------------
extern "C" void kernel_launch(void* const* d_in, const int* in_sizes, int n_in,
                              void* d_out, int out_size, void* d_ws, size_t ws_size,
                              hipStream_t stream)
{
    const float* tok = (const float*)d_in[0];
    const float* pat = (const float*)d_in[1];
    const float* W1  = (const float*)d_in[2];
    const float* b1  = (const float*)d_in[3];
    const float* W2  = (const float*)d_in[4];
    const float* b2  = (const float*)d_in[5];
    float* out = (float*)d_out;

    const int Mtok = 2 * 512;   // 1024
    const int Mpat = 2 * 196;   // 392

    float* tokp = (float*)d_ws;
    float* patp = tokp + (size_t)Mtok * EMBED;

    dim3 wave(32);
    dim3 g_tok(EMBED / 32, (Mtok + 31) / 32);   // 24 x 32
    dim3 g_pat(EMBED / 32, (Mpat + 31) / 32);   // 24 x 13

    WPATask_proj_gemm_wmma<<<g_tok, wave, 0, stream>>>(
        tok, W1, b1, tokp, Mtok, /*wcol_off=*/0,     /*add_bias=*/1);
    WPATask_proj_gemm_wmma<<<g_pat, wave, 0, stream>>>(
        pat, W1, b1, patp, Mpat, /*wcol_off=*/EMBED, /*add_bias=*/0);

    WPATask_score_kernel<<<dim3(2 * 512 / TB), dim3(256), 0, stream>>>(
        tokp, patp, W2, b2, out);
}